// model_52089363366196
// MI455X (gfx1250) — compile-verified
//
#include <hip/hip_runtime.h>
#include <hip/hip_bf16.h>

// ---------------------------------------------------------------------------
// GIN inference on MI455X (gfx1250): fused WMMA f32 GEMMs + atomic scatter.
// ---------------------------------------------------------------------------

#define NNODES   100000
#define NEDGES   1000000
#define HD       128
#define NGRAPHS  256
#define LDS_STRIDE 132   // 128 + 4 pad: bank-conflict-free column reads

typedef __attribute__((ext_vector_type(2))) float v2f;
typedef __attribute__((ext_vector_type(8))) float v8f;

// One wave computes a 16(M) x 16(N) f32 tile: C = A[16x128] * B[128x16] + cinit.
// A comes from LDS (row-major, LDS_STRIDE), B from global (HD x HD row-major,
// columns nbase..nbase+15). Uses V_WMMA_F32_16X16X4_F32, 32 MACs over K=128.
__device__ __forceinline__ v8f wmma_row16(const float* As,
                                          const float* __restrict__ B,
                                          float cinit, int nbase, int lane) {
  v8f c;
#pragma unroll
  for (int i = 0; i < 8; ++i) c[i] = cinit;
  const int m  = lane & 15;
  const int kh = (lane >> 4) << 1;        // lanes 0-15: K+0/1, lanes 16-31: K+2/3
  const int nn = nbase + (lane & 15);
#pragma unroll
  for (int k0 = 0; k0 < HD; k0 += 4) {
    const int kk = k0 + kh;
    v2f a, b;
    a.x = As[m * LDS_STRIDE + kk];
    a.y = As[m * LDS_STRIDE + kk + 1];
    b.x = B[kk * HD + nn];
    b.y = B[(kk + 1) * HD + nn];
    c = __builtin_amdgcn_wmma_f32_16x16x4_f32(false, a, false, b,
                                              (short)0, c, false, false);
  }
  return c;
}

// ---------------------------------------------------------------------------
// Edge aggregation: agg[dst] += h[src].  One wave per edge, float4 per lane.
// ---------------------------------------------------------------------------
__global__ __launch_bounds__(256) void scatter_add_kernel(
    const float* __restrict__ h, const long long* __restrict__ ei,
    float* __restrict__ agg) {
  const int e = (int)((blockIdx.x * 256u + threadIdx.x) >> 5);
  if (e >= NEDGES) return;
  const int lane = threadIdx.x & 31;
  const long long s = ei[e];
  const long long d = ei[NEDGES + e];
  const float4 v = *(const float4*)(h + s * HD + (lane << 2));
  float* ap = agg + d * HD + (lane << 2);
  atomicAdd(ap + 0, v.x);
  atomicAdd(ap + 1, v.y);
  atomicAdd(ap + 2, v.z);
  atomicAdd(ap + 3, v.w);
}

// ---------------------------------------------------------------------------
// Fused GIN layer: hout = bn(relu( relu((hin+agg)@Wa+ba) @ Wb + bb ))
// One block = 16 node rows; 8 waves cover the 128 output columns.
// ---------------------------------------------------------------------------
__global__ __launch_bounds__(256) void gin_layer_kernel(
    const float* __restrict__ hin, const float* __restrict__ agg,
    const float* __restrict__ Wa, const float* __restrict__ ba,
    const float* __restrict__ Wb, const float* __restrict__ bb,
    const float* __restrict__ gamma, const float* __restrict__ beta,
    float* __restrict__ hout) {
  __shared__ float As[16 * LDS_STRIDE];
  __shared__ float Zs[16 * LDS_STRIDE];
  const int t    = threadIdx.x;
  const int lane = t & 31;
  const int wave = t >> 5;
  const long long r0 = (long long)blockIdx.x * 16;

  // zin = hin + agg tile -> LDS (2048 floats, 8 per thread, float4 loads)
  {
    const int flat = t * 8;
    const int row  = flat >> 7;
    const int col  = flat & 127;
    const long long g = (r0 + row) * HD + col;
    const float4 u0 = *(const float4*)(hin + g);
    const float4 u1 = *(const float4*)(agg + g);
    const float4 w0 = *(const float4*)(hin + g + 4);
    const float4 w1 = *(const float4*)(agg + g + 4);
    float* dst = As + row * LDS_STRIDE + col;
    dst[0] = u0.x + u1.x; dst[1] = u0.y + u1.y;
    dst[2] = u0.z + u1.z; dst[3] = u0.w + u1.w;
    dst[4] = w0.x + w1.x; dst[5] = w0.y + w1.y;
    dst[6] = w0.z + w1.z; dst[7] = w0.w + w1.w;
  }
  __syncthreads();

  const int nbase = wave * 16;
  const int nn    = nbase + (lane & 15);
  const int mo    = (lane >> 4) << 3;     // D-matrix: M = mo + r

  // GEMM1 + bias + ReLU -> Zs
  v8f c1 = wmma_row16(As, Wa, ba[nn], nbase, lane);
#pragma unroll
  for (int r = 0; r < 8; ++r) {
    float z = c1[r];
    Zs[(mo + r) * LDS_STRIDE + nn] = z > 0.f ? z : 0.f;
  }
  __syncthreads();

  // GEMM2 + bias + ReLU + eval-BatchNorm -> hout
  v8f c2 = wmma_row16(Zs, Wb, bb[nn], nbase, lane);
  const float scale = gamma[nn] * rsqrtf(1.0f + 1e-5f);
  const float shift = beta[nn];
#pragma unroll
  for (int r = 0; r < 8; ++r) {
    float z = c2[r];
    z = z > 0.f ? z : 0.f;
    hout[(r0 + mo + r) * HD + nn] = z * scale + shift;
  }
}

// ---------------------------------------------------------------------------
// Global add-pool: pooled[batch[n]] += h[n].  One wave per node.
// ---------------------------------------------------------------------------
__global__ __launch_bounds__(256) void pool_kernel(
    const float* __restrict__ h, const long long* __restrict__ batch,
    float* __restrict__ pooled) {
  const int nidx = (int)((blockIdx.x * 256u + threadIdx.x) >> 5);
  if (nidx >= NNODES) return;
  const int lane = threadIdx.x & 31;
  const long long g = batch[nidx];
  const float4 v = *(const float4*)(h + (long long)nidx * HD + (lane << 2));
  float* pp = pooled + g * HD + (lane << 2);
  atomicAdd(pp + 0, v.x);
  atomicAdd(pp + 1, v.y);
  atomicAdd(pp + 2, v.z);
  atomicAdd(pp + 3, v.w);
}

// ---------------------------------------------------------------------------
// Head: emb = relu(pooled@Wf1+bf1) -> out[0:G*HD]; score = emb@Wf2+bf2 -> tail
// ---------------------------------------------------------------------------
__global__ __launch_bounds__(256) void final_mlp_kernel(
    const float* __restrict__ pooled,
    const float* __restrict__ Wf1, const float* __restrict__ bf1,
    const float* __restrict__ Wf2, const float* __restrict__ bf2,
    float* __restrict__ out) {
  __shared__ float As[16 * LDS_STRIDE];
  __shared__ float Zs[16 * LDS_STRIDE];
  const int t    = threadIdx.x;
  const int lane = t & 31;
  const int wave = t >> 5;
  const int r0   = blockIdx.x * 16;
  {
    const int flat = t * 8;
    const int row  = flat >> 7;
    const int col  = flat & 127;
    const float4 u = *(const float4*)(pooled + (r0 + row) * HD + col);
    const float4 w = *(const float4*)(pooled + (r0 + row) * HD + col + 4);
    float* dst = As + row * LDS_STRIDE + col;
    dst[0] = u.x; dst[1] = u.y; dst[2] = u.z; dst[3] = u.w;
    dst[4] = w.x; dst[5] = w.y; dst[6] = w.z; dst[7] = w.w;
  }
  __syncthreads();

  const int nbase = wave * 16;
  const int nn    = nbase + (lane & 15);
  const int mo    = (lane >> 4) << 3;

  v8f c = wmma_row16(As, Wf1, bf1[nn], nbase, lane);
#pragma unroll
  for (int r = 0; r < 8; ++r) {
    float e = c[r];
    e = e > 0.f ? e : 0.f;
    Zs[(mo + r) * LDS_STRIDE + nn] = e;
    out[(r0 + mo + r) * HD + nn]   = e;
  }
  __syncthreads();

  if (t < 16) {
    float acc = bf2[0];
#pragma unroll 4
    for (int j = 0; j < HD; ++j) acc += Zs[t * LDS_STRIDE + j] * Wf2[j];
    out[NGRAPHS * HD + r0 + t] = acc;
  }
}

// ---------------------------------------------------------------------------
extern "C" void kernel_launch(void* const* d_in, const int* in_sizes, int n_in,
                              void* d_out, int out_size, void* d_ws, size_t ws_size,
                              hipStream_t stream) {
  const float*     x     = (const float*)d_in[0];
  const long long* ei    = (const long long*)d_in[1];
  const long long* batch = (const long long*)d_in[2];
  const float* W1a = (const float*)d_in[3];  const float* b1a = (const float*)d_in[4];
  const float* W1b = (const float*)d_in[5];  const float* b1b = (const float*)d_in[6];
  const float* g1  = (const float*)d_in[7];  const float* be1 = (const float*)d_in[8];
  const float* W2a = (const float*)d_in[9];  const float* b2a = (const float*)d_in[10];
  const float* W2b = (const float*)d_in[11]; const float* b2b = (const float*)d_in[12];
  const float* g2  = (const float*)d_in[13]; const float* be2 = (const float*)d_in[14];
  const float* Wf1 = (const float*)d_in[15]; const float* bf1 = (const float*)d_in[16];
  const float* Wf2 = (const float*)d_in[17]; const float* bf2 = (const float*)d_in[18];
  float* out = (float*)d_out;

  const size_t NH = (size_t)NNODES * HD;      // 12.8M floats
  float* agg    = (float*)d_ws;
  float* h1     = agg + NH;
  float* h2     = h1 + NH;
  float* pooled = h2 + NH;

  const int scatterBlocks = (NEDGES * 32 + 255) / 256;   // 125000
  const int ginBlocks     = NNODES / 16;                 // 6250 (exact)
  const int poolBlocks    = (NNODES * 32 + 255) / 256;   // 12500
  const int headBlocks    = NGRAPHS / 16;                // 16 (exact)

  // Layer 1
  hipMemsetAsync(agg, 0, NH * sizeof(float), stream);
  scatter_add_kernel<<<scatterBlocks, 256, 0, stream>>>(x, ei, agg);
  gin_layer_kernel<<<ginBlocks, 256, 0, stream>>>(x, agg, W1a, b1a, W1b, b1b,
                                                  g1, be1, h1);
  // Layer 2
  hipMemsetAsync(agg, 0, NH * sizeof(float), stream);
  scatter_add_kernel<<<scatterBlocks, 256, 0, stream>>>(h1, ei, agg);
  gin_layer_kernel<<<ginBlocks, 256, 0, stream>>>(h1, agg, W2a, b2a, W2b, b2b,
                                                  g2, be2, h2);
  // Pool + head
  hipMemsetAsync(pooled, 0, (size_t)NGRAPHS * HD * sizeof(float), stream);
  pool_kernel<<<poolBlocks, 256, 0, stream>>>(h2, batch, pooled);
  final_mlp_kernel<<<headBlocks, 256, 0, stream>>>(pooled, Wf1, bf1, Wf2, bf2, out);
}